// ReadHead_41515153883272
// MI455X (gfx1250) — compile-verified
//
#include <hip/hip_runtime.h>
#include <hip/hip_bf16.h>
#include <math.h>

// Problem constants (from reference)
#define BATCH 128
#define CTRL  1024
#define NN    4096
#define MM    64
#define OCOLS 70          // M + 6
#define EPSF  1e-16f

typedef __attribute__((ext_vector_type(2))) float v2f;
typedef __attribute__((ext_vector_type(8))) float v8f;

// ---------------------------------------------------------------------------
// Kernel 1: o = emb(128x1024) @ fc_w.T(1024x70) + fc_b  via V_WMMA_F32_16X16X4_F32
// One wave (32 lanes) per 16x16 output tile. 8 row-tiles x 5 col-tiles = 40 waves.
//
// A (16x4 f32) lane layout: lane L (L<16): M=L, VGPR v -> K=v ; lane L>=16: K=v+2
// B (4x16 f32) lane layout: VGPR v, lanes 0-15 -> K=v, N=lane ; lanes 16-31 -> K=v+2
// D (16x16 f32): VGPR v, lanes 0-15 -> M=v, N=lane ; lanes 16-31 -> M=v+8
// ---------------------------------------------------------------------------
__global__ __launch_bounds__(32) void fc_wmma_kernel(
    const float* __restrict__ emb,   // 128 x 1024
    const float* __restrict__ fc_w,  // 70 x 1024
    const float* __restrict__ fc_b,  // 70
    float* __restrict__ o)           // 128 x 70
{
    const int tile = blockIdx.x;          // 0..39
    const int bi = tile / 5;              // row tile (batches)
    const int ji = tile % 5;              // col tile (outputs)
    const int b0 = bi * 16;
    const int j0 = ji * 16;

    const int lane = threadIdx.x;         // 0..31 (wave32)
    const int half = lane >> 4;           // 0 | 1
    const int l16  = lane & 15;
    const int koff = half * 2;            // K offset this lane holds

    const int rowA = b0 + l16;            // batch row for A
    const int j    = j0 + l16;            // output column for B
    const int jc   = (j < OCOLS) ? j : (OCOLS - 1);  // clamp loads, mask stores

    const float* pa = emb  + (size_t)rowA * CTRL + koff;
    const float* pb = fc_w + (size_t)jc   * CTRL + koff;

    v8f acc = {};
    #pragma unroll 4
    for (int c = 0; c < CTRL; c += 4) {
        v2f a = *(const v2f*)(pa + c);    // A[M=rowA, K=c+koff .. c+koff+1]
        v2f bm = *(const v2f*)(pb + c);   // B[K=c+koff.., N=j] = fc_w[j, c+koff..]
        acc = __builtin_amdgcn_wmma_f32_16x16x4_f32(
            /*neg_a=*/false, a, /*neg_b=*/false, bm,
            /*c_mod=*/(short)0, acc, /*reuse_a=*/false, /*reuse_b=*/false);
    }

    if (j < OCOLS) {
        const float bias = fc_b[jc];
        #pragma unroll
        for (int v = 0; v < 8; ++v) {
            const int row = b0 + v + half * 8;
            o[(size_t)row * OCOLS + j] = acc[v] + bias;
        }
    }
}

// ---------------------------------------------------------------------------
// Helpers: block-wide reductions (256 threads = 8 wave32)
// ---------------------------------------------------------------------------
__device__ inline float block_reduce_sum(float v, float* red) {
    #pragma unroll
    for (int off = 16; off > 0; off >>= 1) v += __shfl_down(v, off, 32);
    const int lane = threadIdx.x & 31, wid = threadIdx.x >> 5;
    if (lane == 0) red[wid] = v;
    __syncthreads();
    if (wid == 0) {
        v = (lane < 8) ? red[lane] : 0.0f;
        #pragma unroll
        for (int off = 4; off > 0; off >>= 1) v += __shfl_down(v, off, 32);
        if (lane == 0) red[0] = v;
    }
    __syncthreads();
    v = red[0];
    __syncthreads();
    return v;
}

__device__ inline float block_reduce_max(float v, float* red) {
    #pragma unroll
    for (int off = 16; off > 0; off >>= 1) v = fmaxf(v, __shfl_down(v, off, 32));
    const int lane = threadIdx.x & 31, wid = threadIdx.x >> 5;
    if (lane == 0) red[wid] = v;
    __syncthreads();
    if (wid == 0) {
        v = (lane < 8) ? red[lane] : -INFINITY;
        #pragma unroll
        for (int off = 4; off > 0; off >>= 1) v = fmaxf(v, __shfl_down(v, off, 32));
        if (lane == 0) red[0] = v;
    }
    __syncthreads();
    v = red[0];
    __syncthreads();
    return v;
}

__device__ inline float softplusf(float x) {
    // numerically stable log1p(exp(x))
    return fmaxf(x, 0.0f) + log1pf(expf(-fabsf(x)));
}

// ---------------------------------------------------------------------------
// Kernel 2: one workgroup (256 threads) per batch row.
//   Pass A: fused dot + row-norm over memory[b] (single global read)
//   LDS:    softmax over N, interpolation, circular shift, sharpen, renorm
//   Pass B: readout r = w^T memory[b] (second global read, L2-hot)
// ---------------------------------------------------------------------------
__global__ __launch_bounds__(256) void addressing_kernel(
    const float* __restrict__ o,       // 128 x 70
    const float* __restrict__ w_prev,  // 128 x 4096
    const float* __restrict__ memory,  // 128 x 4096 x 64
    float* __restrict__ out_r,         // 128 x 64
    float* __restrict__ out_w)         // 128 x 4096
{
    __shared__ float k_s[MM];
    __shared__ float w_s[NN];          // holds w_g then final w
    __shared__ float red_s[16];
    __shared__ float par_s[8];         // beta,g,s0,s1,s2,gamma,knorm
    __shared__ float rpart[4][MM];

    const int b   = blockIdx.x;
    const int tid = threadIdx.x;
    const float* ob   = o + (size_t)b * OCOLS;
    const float* memb = memory + (size_t)b * NN * MM;

    // ---- load k, compute |k| and scalar gates --------------------------------
    float ksq = 0.0f;
    if (tid < MM) {
        const float kv = ob[tid];
        k_s[tid] = kv;
        ksq = kv * kv;
    }
    const float knorm2 = block_reduce_sum(ksq, red_s);
    if (tid == 0) {
        par_s[6] = sqrtf(knorm2);                          // k_norm
        par_s[0] = softplusf(ob[MM]);                      // beta
        par_s[1] = 1.0f / (1.0f + expf(-ob[MM + 1]));      // g
        const float e0 = ob[MM + 2], e1 = ob[MM + 3], e2 = ob[MM + 4];
        const float mx = fmaxf(e0, fmaxf(e1, e2));
        const float x0 = expf(e0 - mx), x1 = expf(e1 - mx), x2 = expf(e2 - mx);
        const float sden = x0 + x1 + x2;
        par_s[2] = x0 / sden; par_s[3] = x1 / sden; par_s[4] = x2 / sden;
        par_s[5] = 1.0f + softplusf(ob[MM + 5]);           // gamma
    }
    __syncthreads();

    const float beta  = par_s[0];
    const float g     = par_s[1];
    const float knorm = par_s[6];

    // ---- Pass A: z = beta * cosine_sim, 16 rows per thread -------------------
    float zloc[16];
    #pragma unroll
    for (int i = 0; i < 16; ++i) {
        const int n = tid + i * 256;
        const float* row = memb + (size_t)n * MM;
        float dot = 0.0f, msq = 0.0f;
        #pragma unroll
        for (int m4 = 0; m4 < MM; m4 += 4) {
            const float4 mv = *(const float4*)(row + m4);
            dot += mv.x * k_s[m4]     + mv.y * k_s[m4 + 1]
                 + mv.z * k_s[m4 + 2] + mv.w * k_s[m4 + 3];
            msq += mv.x * mv.x + mv.y * mv.y + mv.z * mv.z + mv.w * mv.w;
        }
        zloc[i] = beta * (dot / (knorm * sqrtf(msq) + EPSF));
    }

    // ---- softmax over N (in registers + block reductions) --------------------
    float lmax = -INFINITY;
    #pragma unroll
    for (int i = 0; i < 16; ++i) lmax = fmaxf(lmax, zloc[i]);
    const float gmax = block_reduce_max(lmax, red_s);

    float lsum = 0.0f;
    #pragma unroll
    for (int i = 0; i < 16; ++i) { zloc[i] = expf(zloc[i] - gmax); lsum += zloc[i]; }
    const float gsum = block_reduce_sum(lsum, red_s);
    const float inv  = 1.0f / gsum;

    // ---- interpolation: w_g into LDS ----------------------------------------
    const float* wpb = w_prev + (size_t)b * NN;
    #pragma unroll
    for (int i = 0; i < 16; ++i) {
        const int n = tid + i * 256;
        const float wc = zloc[i] * inv;
        w_s[n] = g * wc + (1.0f - g) * wpb[n];
    }
    __syncthreads();

    // ---- circular 3-tap shift + sharpening ----------------------------------
    const float s0 = par_s[2], s1 = par_s[3], s2 = par_s[4], gamma = par_s[5];
    float wp[16];
    float lsum2 = 0.0f;
    #pragma unroll
    for (int i = 0; i < 16; ++i) {
        const int n  = tid + i * 256;
        const int nm = (n == 0)      ? NN - 1 : n - 1;
        const int np = (n == NN - 1) ? 0      : n + 1;
        const float wt = s0 * w_s[nm] + s1 * w_s[n] + s2 * w_s[np];
        wp[i] = powf(wt, gamma);
        lsum2 += wp[i];
    }
    const float gsum2 = block_reduce_sum(lsum2, red_s);  // barrier: safe to overwrite w_s
    const float invp  = 1.0f / (gsum2 + EPSF);

    #pragma unroll
    for (int i = 0; i < 16; ++i) {
        const int n = tid + i * 256;
        const float w = wp[i] * invp;
        w_s[n] = w;
        out_w[(size_t)b * NN + n] = w;
    }
    __syncthreads();

    // ---- Pass B: readout r[m] = sum_n w[n] * memory[b,n,m] -------------------
    const int m   = tid & 63;
    const int grp = tid >> 6;          // 0..3
    float acc = 0.0f;
    for (int n = grp; n < NN; n += 4) {
        if (n + 16 < NN)
            __builtin_prefetch(memb + (size_t)(n + 16) * MM + m, 0, 1);
        acc += w_s[n] * memb[(size_t)n * MM + m];
    }
    rpart[grp][m] = acc;
    __syncthreads();
    if (tid < MM) {
        out_r[(size_t)b * MM + tid] =
            rpart[0][tid] + rpart[1][tid] + rpart[2][tid] + rpart[3][tid];
    }
}

// ---------------------------------------------------------------------------
// Launcher
// ---------------------------------------------------------------------------
extern "C" void kernel_launch(void* const* d_in, const int* in_sizes, int n_in,
                              void* d_out, int out_size, void* d_ws, size_t ws_size,
                              hipStream_t stream) {
    const float* emb    = (const float*)d_in[0];   // 128*1024
    const float* w_prev = (const float*)d_in[1];   // 128*4096
    const float* memory = (const float*)d_in[2];   // 128*4096*64
    const float* fc_w   = (const float*)d_in[3];   // 70*1024
    const float* fc_b   = (const float*)d_in[4];   // 70

    float* out   = (float*)d_out;                  // r (128*64) then w (128*4096)
    float* out_r = out;
    float* out_w = out + BATCH * MM;

    float* o = (float*)d_ws;                       // 128*70 f32 scratch

    fc_wmma_kernel<<<dim3(40), dim3(32), 0, stream>>>(emb, fc_w, fc_b, o);
    addressing_kernel<<<dim3(BATCH), dim3(256), 0, stream>>>(o, w_prev, memory, out_r, out_w);
}